// DocREModel_23201413333115
// MI455X (gfx1250) — compile-verified
//
#include <hip/hip_runtime.h>
#include <math.h>

typedef __attribute__((ext_vector_type(2))) float v2f;
typedef __attribute__((ext_vector_type(8))) float v8f;

#define B_    4
#define L_    1024
#define H_    768
#define NH_   12
#define E_    30
#define M_    8
#define P_    512
#define NLAB  97
#define NPAIR (B_*P_)   /* 2048 */

// ---------------------------------------------------------------------------
// ent_emb[b,e,h] = logsumexp_m seq[b, mention_idx[b,e,m]+1, h]
__global__ void k_ent_emb(const float* __restrict__ seq, const int* __restrict__ mi,
                          float* __restrict__ ent_emb) {
  int idx = blockIdx.x * blockDim.x + threadIdx.x;
  if (idx >= B_ * E_ * H_) return;
  int h = idx % H_, be = idx / H_;
  int b = be / E_;
  const int* mp = mi + be * M_;
  float vals[M_]; float mx = -1e30f;
  #pragma unroll
  for (int m = 0; m < M_; ++m) {
    int pos = mp[m] + 1;
    float v = seq[((size_t)(b * L_ + pos)) * H_ + h];
    vals[m] = v; mx = fmaxf(mx, v);
  }
  float s = 0.f;
  #pragma unroll
  for (int m = 0; m < M_; ++m) s += __expf(vals[m] - mx);
  ent_emb[idx] = mx + __logf(s);
}

// ent_att[b,e,nh,l] = mean_m attn[b,nh,pos,l]
__global__ void k_ent_att(const float* __restrict__ attn, const int* __restrict__ mi,
                          float* __restrict__ ent_att) {
  int idx = blockIdx.x * blockDim.x + threadIdx.x;
  if (idx >= B_ * E_ * NH_ * L_) return;
  int l = idx % L_;
  int r = idx / L_;
  int nh = r % NH_;
  int be = r / NH_;
  int b = be / E_;
  const int* mp = mi + be * M_;
  float s = 0.f;
  #pragma unroll
  for (int m = 0; m < M_; ++m) {
    int pos = mp[m] + 1;
    s += attn[(((size_t)(b * NH_ + nh)) * L_ + pos) * L_ + l];
  }
  ent_att[idx] = s * (1.f / M_);
}

// hs/ts gather from ent_emb via ht_pairs
__global__ void k_gather_ht(const float* __restrict__ ent_emb, const int* __restrict__ htp,
                            float* __restrict__ hs, float* __restrict__ ts) {
  int idx = blockIdx.x * blockDim.x + threadIdx.x;
  if (idx >= NPAIR * H_) return;
  int h = idx % H_, n = idx / H_;
  int b = n / P_;
  int hi = htp[n * 2 + 0], ti = htp[n * 2 + 1];
  hs[idx] = ent_emb[(size_t)(b * E_ + hi) * H_ + h];
  ts[idx] = ent_emb[(size_t)(b * E_ + ti) * H_ + h];
}

// ht_att[b,p,l] = mean_nh(ea[h]*ea[t]) then normalize over l
__global__ __launch_bounds__(256) void k_ht_att(const float* __restrict__ ent_att,
                                                const int* __restrict__ htp,
                                                float* __restrict__ ht_att) {
  __shared__ float red[256];
  int bp = blockIdx.x;
  int b = bp / P_;
  int tid = threadIdx.x;
  int hi = htp[bp * 2 + 0], ti = htp[bp * 2 + 1];
  const float* ah = ent_att + (size_t)(b * E_ + hi) * NH_ * L_;
  const float* at = ent_att + (size_t)(b * E_ + ti) * NH_ * L_;
  float* o = ht_att + (size_t)bp * L_;
  float lsum = 0.f;
  for (int l = tid; l < L_; l += 256) {
    float s = 0.f;
    #pragma unroll
    for (int nh = 0; nh < NH_; ++nh) s += ah[nh * L_ + l] * at[nh * L_ + l];
    s *= (1.f / NH_);
    o[l] = s; lsum += s;
  }
  red[tid] = lsum; __syncthreads();
  for (int s = 128; s > 0; s >>= 1) { if (tid < s) red[tid] += red[tid + s]; __syncthreads(); }
  float inv = 1.f / (red[0] + 1e-5f);
  for (int l = tid; l < L_; l += 256) o[l] *= inv;
}

// ---------------------------------------------------------------------------
// Generic guarded WMMA fp32 GEMM: C = act(A_eff @ B_eff + bias)
// A_eff row r, col k: k<ksplit -> A1[r*lda1+k] else A2[r*lda2+(k-ksplit)]
// B_eff[k,n]: btrans ? B[n*ldb+k] : B[k*ldb+n].  64x64 block tile, 4 waves.
__global__ __launch_bounds__(128) void k_gemm(
    const float* __restrict__ A1, long sA, int lda1,
    const float* __restrict__ A2, int lda2, int ksplit,
    const float* __restrict__ Bm, long sB, int ldb, int btrans,
    float* __restrict__ C, long sC, int ldc,
    const float* __restrict__ bias, int M, int N, int K, int act)
{
  __shared__ float As[64 * 18];
  __shared__ float Bs[64 * 18];
  const int tid  = threadIdx.x;
  const int lane = tid & 31;
  const int wv   = tid >> 5;
  const int m0   = blockIdx.y * 64;
  const int n0   = blockIdx.x * 64;
  const float* a1 = A1 + (size_t)blockIdx.z * sA;
  const float* bm = Bm + (size_t)blockIdx.z * sB;
  float* cc = C + (size_t)blockIdx.z * sC;

  v8f acc[4];
  v8f zero = {0.f,0.f,0.f,0.f,0.f,0.f,0.f,0.f};
  for (int i = 0; i < 4; ++i) acc[i] = zero;

  const int mi = lane & 15;
  const int k0 = (lane >> 4) << 1;

  for (int kc = 0; kc < K; kc += 16) {
    for (int t = tid; t < 64 * 16; t += 128) {
      int m = t >> 4, k = t & 15;
      int gk = kc + k, gm = m0 + m;
      float v = 0.f;
      if (gm < M && gk < K)
        v = (gk < ksplit) ? a1[(size_t)gm * lda1 + gk]
                          : A2[(size_t)gm * lda2 + (gk - ksplit)];
      As[m * 18 + k] = v;
    }
    for (int t = tid; t < 64 * 16; t += 128) {
      int n = t >> 4, k = t & 15;
      int gk = kc + k, gn = n0 + n;
      float v = 0.f;
      if (gn < N && gk < K)
        v = btrans ? bm[(size_t)gn * ldb + gk] : bm[(size_t)gk * ldb + gn];
      Bs[n * 18 + k] = v;  // k-transposed stage -> contiguous k-pairs per n
    }
    __syncthreads();
    #pragma unroll
    for (int kk = 0; kk < 16; kk += 4) {
      v2f af = *(const v2f*)&As[(wv * 16 + mi) * 18 + kk + k0];
      #pragma unroll
      for (int nt = 0; nt < 4; ++nt) {
        v2f bf = *(const v2f*)&Bs[(nt * 16 + mi) * 18 + kk + k0];
        acc[nt] = __builtin_amdgcn_wmma_f32_16x16x4_f32(
            false, af, false, bf, (short)0, acc[nt], false, false);
      }
    }
    __syncthreads();
  }
  const int hi8 = (lane >> 4) << 3;
  for (int nt = 0; nt < 4; ++nt) {
    int gn = n0 + nt * 16 + mi;
    if (gn >= N) continue;
    float bv = bias ? bias[gn] : 0.f;
    #pragma unroll
    for (int i = 0; i < 8; ++i) {
      int gm = m0 + wv * 16 + i + hi8;
      if (gm < M) {
        float v = acc[nt][i] + bv;
        if (act == 1) v = tanhf(v);
        cc[(size_t)gm * ldc + gn] = v;
      }
    }
  }
}

// ---------------------------------------------------------------------------
// GAT helper: el[v,h] = sum_d z[v,h,d]*al[h,d]; er likewise
__global__ void k_el_er(const float* __restrict__ z, const float* __restrict__ al,
                        const float* __restrict__ ar, float* __restrict__ el,
                        float* __restrict__ er, int nh, int od) {
  int t = blockIdx.x * blockDim.x + threadIdx.x;
  if (t >= NLAB * nh) return;
  int v = t / nh, h = t % nh;
  const float* zr = z + (size_t)(v * nh + h) * od;
  const float* alr = al + h * od;
  const float* arr = ar + h * od;
  float sl = 0.f, sr = 0.f;
  for (int d = 0; d < od; ++d) { float zz = zr[d]; sl += zz * alr[d]; sr += zz * arr[d]; }
  el[t] = sl; er[t] = sr;
}

// GAT aggregate: out[v,h,d] = sum_u softmax_u(lrelu(er[v]+el[u]) masked adj[u,v]) * z[u,h,d]
__global__ __launch_bounds__(256) void k_gat_agg(
    const float* __restrict__ z, const float* __restrict__ el,
    const float* __restrict__ er, const int* __restrict__ adj,
    float* __restrict__ out, int nh, int od, int do_elu)
{
  int v = blockIdx.x, h = blockIdx.y;
  int tid = threadIdx.x;
  float erv = er[v * nh + h];
  float mx = -1e30f;
  for (int u = 0; u < NLAB; ++u) {
    if (adj[u * NLAB + v]) {
      float e = erv + el[u * nh + h];
      e = (e >= 0.f) ? e : 0.2f * e;
      mx = fmaxf(mx, e);
    }
  }
  float den = 0.f;
  for (int u = 0; u < NLAB; ++u) {
    if (adj[u * NLAB + v]) {
      float e = erv + el[u * nh + h];
      e = (e >= 0.f) ? e : 0.2f * e;
      den += __expf(e - mx);
    }
  }
  float inv = 1.f / den;
  float accv[4] = {0.f, 0.f, 0.f, 0.f};
  for (int u = 0; u < NLAB; ++u) {
    if (!adj[u * NLAB + v]) continue;
    float e = erv + el[u * nh + h];
    e = (e >= 0.f) ? e : 0.2f * e;
    float a = __expf(e - mx) * inv;
    const float* zr = z + (size_t)(u * nh + h) * od;
    #pragma unroll
    for (int r = 0; r < 4; ++r) {
      int d = tid + (r << 8);
      if (d < od) accv[r] += a * zr[d];
    }
  }
  float* orow = out + (size_t)(v * nh + h) * od;
  #pragma unroll
  for (int r = 0; r < 4; ++r) {
    int d = tid + (r << 8);
    if (d < od) {
      float o = accv[r];
      if (do_elu) o = (o > 0.f) ? o : (__expf(o) - 1.f);
      orow[d] = o;
    }
  }
}

// Row LayerNorm over NLAB classes (in place)
__global__ __launch_bounds__(128) void k_layernorm(float* __restrict__ x,
                                                   const float* __restrict__ g,
                                                   const float* __restrict__ b) {
  __shared__ float red[128];
  float* row = x + (size_t)blockIdx.x * NLAB;
  int tid = threadIdx.x;
  float v = (tid < NLAB) ? row[tid] : 0.f;
  red[tid] = v; __syncthreads();
  for (int s = 64; s > 0; s >>= 1) { if (tid < s) red[tid] += red[tid + s]; __syncthreads(); }
  float mean = red[0] * (1.f / NLAB); __syncthreads();
  float d = (tid < NLAB) ? (v - mean) : 0.f;
  red[tid] = d * d; __syncthreads();
  for (int s = 64; s > 0; s >>= 1) { if (tid < s) red[tid] += red[tid + s]; __syncthreads(); }
  float var = red[0] * (1.f / NLAB);
  if (tid < NLAB) row[tid] = (v - mean) * rsqrtf(var + 1e-5f) * g[tid] + b[tid];
}

// ---------------------------------------------------------------------------
// Fused bilinear: logits[n,c] = sum_{blk,i,j} hs2[n,blk,i]*ts2[n,blk,j]*W[(blk*64+i)*64+j, c]
//                + sum_q lli[n,q]*W[49152+q, c] + b_bil[c]
// 16 rows per block, 7 waves x 16-class column tiles. bl generated in registers.
__global__ __launch_bounds__(224) void k_bilinear(
    const float* __restrict__ hs2, const float* __restrict__ ts2,
    const float* __restrict__ lli, const float* __restrict__ W,
    const float* __restrict__ bb, float* __restrict__ out)
{
  __shared__ float ts2s[16 * 770];   // padded stride: conflict-free b64 reads
  __shared__ float lls[16 * 112];
  const int tid  = threadIdx.x;
  const int row0 = blockIdx.x * 16;
  for (int t = tid; t < 16 * 768; t += 224) {
    int r = t / 768, c = t % 768;
    ts2s[r * 770 + c] = ts2[(size_t)(row0 + r) * 768 + c];
  }
  for (int t = tid; t < 16 * 112; t += 224) {
    int r = t / 112, c = t % 112;
    lls[t] = (c < NLAB) ? lli[(size_t)(row0 + r) * NLAB + c] : 0.f;
  }
  __syncthreads();

  const int lane = tid & 31;
  const int wv   = tid >> 5;          // 0..6 -> class tile
  const int c0   = wv * 16;
  const int mi   = lane & 15;         // row for A-frag, class for B-frag
  const int k0   = (lane >> 4) << 1;  // 0 or 2
  v8f acc = {0.f,0.f,0.f,0.f,0.f,0.f,0.f,0.f};

  const float* hrow = hs2 + (size_t)(row0 + mi) * 768;
  for (int blk = 0; blk < 12; ++blk) {
    const float* tsr = &ts2s[mi * 770 + blk * 64];
    for (int i = 0; i < 64; ++i) {
      float hv = hrow[blk * 64 + i];
      const float* wr = W + (size_t)(blk * 4096 + i * 64 + k0) * NLAB + c0 + mi;
      __builtin_prefetch(wr + 64 * NLAB, 0, 0);   // next i-row of W (L2 hot)
      #pragma unroll
      for (int j0 = 0; j0 < 64; j0 += 4) {
        v2f tv = *(const v2f*)&tsr[j0 + k0];
        v2f af = { hv * tv.x, hv * tv.y };
        v2f bf = { wr[(size_t)j0 * NLAB], wr[(size_t)(j0 + 1) * NLAB] };
        acc = __builtin_amdgcn_wmma_f32_16x16x4_f32(
            false, af, false, bf, (short)0, acc, false, false);
      }
    }
  }
  // tail: logits_li columns of W_bil (rows 49152..49248)
  for (int q0 = 0; q0 < 112; q0 += 4) {
    v2f af = *(const v2f*)&lls[mi * 112 + q0 + k0];
    int r0 = 49152 + q0 + k0;
    int ra = (r0     < 49249) ? r0       : 49248;  // af==0 there, value irrelevant
    int rb = (r0 + 1 < 49249) ? (r0 + 1) : 49248;
    v2f bf = { W[(size_t)ra * NLAB + c0 + mi], W[(size_t)rb * NLAB + c0 + mi] };
    acc = __builtin_amdgcn_wmma_f32_16x16x4_f32(
        false, af, false, bf, (short)0, acc, false, false);
  }
  const int hi8 = (lane >> 4) << 3;
  int gc = c0 + mi;
  if (gc < NLAB) {
    float bv = bb[gc];
    #pragma unroll
    for (int i = 0; i < 8; ++i) {
      int gm = row0 + i + hi8;
      out[(size_t)gm * NLAB + gc] = acc[i] + bv;
    }
  }
}

// ---------------------------------------------------------------------------
extern "C" void kernel_launch(void* const* d_in, const int* in_sizes, int n_in,
                              void* d_out, int out_size, void* d_ws, size_t ws_size,
                              hipStream_t stream)
{
  (void)in_sizes; (void)n_in; (void)out_size; (void)ws_size;
  const float* seq  = (const float*)d_in[0];
  const float* attn = (const float*)d_in[1];
  const int*   mi   = (const int*)d_in[2];
  const int*   htp  = (const int*)d_in[3];
  const int*   adj  = (const int*)d_in[4];
  const float* lemb = (const float*)d_in[5];
  const float* gW0  = (const float*)d_in[6];
  const float* gal0 = (const float*)d_in[7];
  const float* gar0 = (const float*)d_in[8];
  const float* gW1  = (const float*)d_in[9];
  const float* gal1 = (const float*)d_in[10];
  const float* gar1 = (const float*)d_in[11];
  const float* gW2  = (const float*)d_in[12];
  const float* gal2 = (const float*)d_in[13];
  const float* gar2 = (const float*)d_in[14];
  const float* lng  = (const float*)d_in[15];
  const float* lnb  = (const float*)d_in[16];
  const float* Wl2  = (const float*)d_in[17];
  const float* bl2  = (const float*)d_in[18];
  const float* Wh   = (const float*)d_in[19];
  const float* bh   = (const float*)d_in[20];
  const float* Wt   = (const float*)d_in[21];
  const float* btl  = (const float*)d_in[22];
  const float* Wb   = (const float*)d_in[23];
  const float* bb   = (const float*)d_in[24];
  float* out = (float*)d_out;

  float* ws = (float*)d_ws;
  size_t off = 0;
  auto alloc = [&](size_t n) { float* p = ws + off; off += n; return p; };
  float* ent_emb = alloc((size_t)B_ * E_ * H_);
  float* ent_att = alloc((size_t)B_ * E_ * NH_ * L_);
  float* ht_att  = alloc((size_t)B_ * P_ * L_);
  float* rs      = alloc((size_t)NPAIR * H_);
  float* hs      = alloc((size_t)NPAIR * H_);
  float* ts      = alloc((size_t)NPAIR * H_);
  float* hs2     = alloc((size_t)NPAIR * H_);
  float* ts2     = alloc((size_t)NPAIR * H_);
  float* zbuf    = alloc((size_t)NLAB * 1000);
  float* xbuf    = alloc((size_t)NLAB * 1000);
  float* lab     = alloc((size_t)NLAB * H_);
  float* el      = alloc(NLAB * 2);
  float* er      = alloc(NLAB * 2);
  float* hs_li   = alloc((size_t)NPAIR * NLAB);
  float* ts_li   = alloc((size_t)NPAIR * NLAB);
  float* lli     = alloc((size_t)NPAIR * NLAB);

  auto gemm = [&](const float* A1, long sA, int lda1,
                  const float* A2, int lda2, int ksplit,
                  const float* Bm, long sB, int ldb, int btr,
                  float* C, long sC, int ldc, const float* bias,
                  int M, int N, int K, int act, int batch) {
    dim3 g((unsigned)((N + 63) / 64), (unsigned)((M + 63) / 64), (unsigned)batch);
    k_gemm<<<g, 128, 0, stream>>>(A1, sA, lda1, A2, lda2, ksplit,
                                  Bm, sB, ldb, btr, C, sC, ldc, bias, M, N, K, act);
  };

  // entity embeddings / attentions
  k_ent_emb<<<(B_*E_*H_ + 255) / 256, 256, 0, stream>>>(seq, mi, ent_emb);
  k_ent_att<<<(B_*E_*NH_*L_ + 255) / 256, 256, 0, stream>>>(attn, mi, ent_att);
  k_gather_ht<<<(NPAIR*H_ + 255) / 256, 256, 0, stream>>>(ent_emb, htp, hs, ts);
  k_ht_att<<<NPAIR, 256, 0, stream>>>(ent_att, htp, ht_att);
  // rs = ht_att @ seq_out (batched)
  gemm(ht_att, (long)P_*L_, L_, nullptr, 0, L_,
       seq, (long)L_*H_, H_, 0, rs, (long)P_*H_, H_, nullptr,
       P_, H_, L_, 0, B_);

  // Label GAT (3 layers)
  gemm(lemb, 0, H_, nullptr, 0, H_, gW0, 0, 1000, 0, zbuf, 0, 1000, nullptr,
       NLAB, 1000, H_, 0, 1);
  k_el_er<<<1, 256, 0, stream>>>(zbuf, gal0, gar0, el, er, 2, 500);
  k_gat_agg<<<dim3(NLAB, 2), 256, 0, stream>>>(zbuf, el, er, adj, xbuf, 2, 500, 1);
  gemm(xbuf, 0, 1000, nullptr, 0, 1000, gW1, 0, 1000, 0, zbuf, 0, 1000, nullptr,
       NLAB, 1000, 1000, 0, 1);
  k_el_er<<<1, 256, 0, stream>>>(zbuf, gal1, gar1, el, er, 2, 500);
  k_gat_agg<<<dim3(NLAB, 2), 256, 0, stream>>>(zbuf, el, er, adj, xbuf, 2, 500, 1);
  gemm(xbuf, 0, 1000, nullptr, 0, 1000, gW2, 0, H_, 0, zbuf, 0, H_, nullptr,
       NLAB, H_, 1000, 0, 1);
  k_el_er<<<1, 256, 0, stream>>>(zbuf, gal2, gar2, el, er, 1, H_);
  k_gat_agg<<<dim3(NLAB, 1), 256, 0, stream>>>(zbuf, el, er, adj, lab, 1, H_, 0);

  // label-informed logits
  gemm(hs, 0, H_, nullptr, 0, H_, lab, 0, H_, 1, hs_li, 0, NLAB, nullptr,
       NPAIR, NLAB, H_, 0, 1);
  gemm(ts, 0, H_, nullptr, 0, H_, lab, 0, H_, 1, ts_li, 0, NLAB, nullptr,
       NPAIR, NLAB, H_, 0, 1);
  k_layernorm<<<NPAIR, 128, 0, stream>>>(hs_li, lng, lnb);
  k_layernorm<<<NPAIR, 128, 0, stream>>>(ts_li, lng, lnb);
  gemm(hs_li, 0, NLAB, ts_li, NLAB, NLAB, Wl2, 0, NLAB, 0, lli, 0, NLAB, bl2,
       NPAIR, NLAB, 2 * NLAB, 0, 1);

  // head/tail projections with concat-K and tanh epilogue
  gemm(hs, 0, H_, rs, H_, H_, Wh, 0, H_, 0, hs2, 0, H_, bh,
       NPAIR, H_, 2 * H_, 1, 1);
  gemm(ts, 0, H_, rs, H_, H_, Wt, 0, H_, 0, ts2, 0, H_, btl,
       NPAIR, H_, 2 * H_, 1, 1);

  // fused bilinear classifier
  k_bilinear<<<NPAIR / 16, 224, 0, stream>>>(hs2, ts2, lli, Wb, bb, out);
}